// HermiteMemoryTest_80126909874851
// MI455X (gfx1250) — compile-verified
//
#include <hip/hip_runtime.h>
#include <hip/hip_bf16.h>

typedef float v2f __attribute__((ext_vector_type(2)));
typedef float v8f __attribute__((ext_vector_type(8)));

#define EPS     1e-6f
#define KDIM    32
#define HDIM    6
#define ODIM    31
#define QDIM    192              // K*H (GEMM inner dim)
#define ROWS    128              // M rows per workgroup
#define QSTRIDE 196              // LDS row stride: 196 % 64 == 4 -> conflict-free b64 frag loads
#define NPAD    32               // O padded to 32

__global__ __launch_bounds__(256) void hermite_wmma_f32_kernel(
    const float* __restrict__ mlp,     // [M,31]
    const float* __restrict__ coord,   // [M,32,2]
    const float* __restrict__ sigmas,  // [32]
    const float* __restrict__ gw,      // [M,32]
    const float* __restrict__ W,       // [32,6,31] == [q=192][o=31]
    float* __restrict__ out)           // [M,31]
{
    extern __shared__ float smem[];
    float* Alds = smem;                    // [ROWS][QSTRIDE]
    float* Wt   = smem + ROWS * QSTRIDE;   // [NPAD][QSTRIDE], Wt[n][q] = W[q][n]

    const int tid   = threadIdx.x;
    const int mbase = blockIdx.x * ROWS;

    // ---- Phase 1a: stage rbf_weights transposed into LDS (zero-pad column o=31) ----
    for (int idx = tid; idx < NPAD * QDIM; idx += 256) {
        const int o = idx / QDIM;
        const int q = idx - o * QDIM;
        Wt[o * QSTRIDE + q] = (o < ODIM) ? W[q * ODIM + o] : 0.0f;
    }

    // ---- Phase 1b: A[r][k*6+h] = gw * hermite_coeff into LDS ----
    {
        const int   k      = tid & 31;          // one k per thread (coalesced)
        const float s      = sigmas[k];
        const float s2     = s * s;
        const float inv_s2 = 1.0f / (s2 + EPS);
        const float inv_s4 = 1.0f / (s2 * s2 + EPS);
        const int   r0     = tid >> 5;
        #pragma unroll
        for (int i = 0; i < 16; ++i) {
            const int r = r0 + i * 8;
            const int m = mbase + r;
            const v2f d = *reinterpret_cast<const v2f*>(&coord[(size_t)(m * KDIM + k) * 2]);
            const float dx = d.x, dy = d.y;
            const float g  = gw[(size_t)m * KDIM + k];
            float* arow = &Alds[r * QSTRIDE + k * HDIM];
            arow[0] = g;                                  // h=0: 1
            arow[1] = g * (-dy * inv_s2);                 // h=1
            arow[2] = g * (-dx * inv_s2);                 // h=2
            arow[3] = g * (dy * dy * inv_s4 - inv_s2);    // h=3
            arow[4] = g * (dx * dy * inv_s4);             // h=4
            arow[5] = g * (dx * dx * inv_s4 - inv_s2);    // h=5
        }
    }

    __syncthreads();

    // ---- Phase 2: FP32 WMMA 16x16x4, K = 192 (48 steps), 2 N-tiles ----
    const int lane  = tid & 31;
    const int w     = tid >> 5;          // wave id; rows [w*16, w*16+16)
    const int row16 = lane & 15;
    const int koff  = (lane >> 4) << 1;  // lanes 16-31 hold K+2,K+3 (ISA 16x4 A layout)

    const int a_base  = (w * 16 + row16) * QSTRIDE + koff;
    const int b_base0 = row16 * QSTRIDE + koff;          // N-tile 0 (o = 0..15)
    const int b_base1 = (16 + row16) * QSTRIDE + koff;   // N-tile 1 (o = 16..31)

    v8f acc0 = {};
    v8f acc1 = {};
    #pragma unroll
    for (int s = 0; s < QDIM / 4; ++s) {
        const int q0 = s * 4;
        const v2f av  = *reinterpret_cast<const v2f*>(&Alds[a_base  + q0]);
        const v2f bv0 = *reinterpret_cast<const v2f*>(&Wt  [b_base0 + q0]);
        const v2f bv1 = *reinterpret_cast<const v2f*>(&Wt  [b_base1 + q0]);
        acc0 = __builtin_amdgcn_wmma_f32_16x16x4_f32(false, av, false, bv0,
                                                     (short)0, acc0, false, false);
        acc1 = __builtin_amdgcn_wmma_f32_16x16x4_f32(false, av, false, bv1,
                                                     (short)0, acc1, false, false);
    }

    // ---- Phase 3: out = mlp * mix, using C/D layout (VGPR r -> M=r / M=8+r) ----
    const int mrow_base = mbase + w * 16 + ((lane < 16) ? 0 : 8);
    const int ocol      = lane & 15;
    #pragma unroll
    for (int r = 0; r < 8; ++r) {
        const size_t base = (size_t)(mrow_base + r) * ODIM;
        out[base + ocol] = mlp[base + ocol] * acc0[r];
        const int o1 = ocol + 16;
        if (o1 < ODIM)                       // skip padded o == 31
            out[base + o1] = mlp[base + o1] * acc1[r];
    }
}

extern "C" void kernel_launch(void* const* d_in, const int* in_sizes, int n_in,
                              void* d_out, int out_size, void* d_ws, size_t ws_size,
                              hipStream_t stream) {
    const float* mlp    = (const float*)d_in[0];  // [B,N,31]
    const float* coord  = (const float*)d_in[1];  // [B,N,32,2]
    const float* sigmas = (const float*)d_in[2];  // [32]
    const float* gw     = (const float*)d_in[3];  // [B,N,32]
    const float* W      = (const float*)d_in[4];  // [32,6,31]
    float* out = (float*)d_out;

    const int M      = in_sizes[0] / ODIM;            // 262144
    const int blocks = M / ROWS;                      // 2048
    const size_t lds = (size_t)(ROWS + NPAD) * QSTRIDE * sizeof(float); // 125,440 B

    hermite_wmma_f32_kernel<<<blocks, 256, lds, stream>>>(mlp, coord, sigmas, gw, W, out);
}